// GCN3_85512798863823
// MI455X (gfx1250) — compile-verified
//
#include <hip/hip_runtime.h>

// GCN 3-layer forward for MI455X (gfx1250, wave32).
// GEMMs use native fp32 WMMA (v_wmma_f32_16x16x4_f32); sparse aggregation is
// edge-parallel f32 atomics (operands are L2-resident: 51MB << 192MB L2).

typedef float v2f __attribute__((ext_vector_type(2)));
typedef float v8f __attribute__((ext_vector_type(8)));

#define WAVES_PER_BLOCK 5
#define GEMM_THREADS (WAVES_PER_BLOCK * 32)

// ---------------------------------------------------------------------------
// WMMA GEMM: H[M,NOUT] = X[M,K] @ W[K,NOUT], row-major, fp32.
// One 16-row m-tile (16 x NOUT outputs) per wave; wave keeps accumulators for ALL n-tiles so
// the LDS-staged K-slab of W and the A fragments are reused across N.
// ---------------------------------------------------------------------------
template <int K, int NOUT>
__launch_bounds__(GEMM_THREADS)
__global__ void gemm_wmma_f32(const float* __restrict__ X,
                              const float* __restrict__ W,
                              float* __restrict__ H, int mtiles) {
  constexpr int KSLAB = 32;
  constexpr int NT = NOUT / 16;
  __shared__ float sW[KSLAB * NOUT];

  const int wave = threadIdx.x >> 5;
  const int lane = threadIdx.x & 31;
  const int half = lane >> 4;   // 0: lanes 0-15, 1: lanes 16-31
  const int lm   = lane & 15;

  int mt = blockIdx.x * WAVES_PER_BLOCK + wave;
  const bool valid = (mt < mtiles);
  if (!valid) mt = mtiles - 1;            // keep EXEC full; skip store later
  const int m0 = mt * 16;
  const float* xrow = X + (size_t)(m0 + lm) * K;

  v8f acc[NT] = {};

  for (int ks = 0; ks < K; ks += KSLAB) {
    // Cooperative load of W[ks .. ks+KSLAB) rows (contiguous) into LDS.
    const float4* wsrc = (const float4*)(W + (size_t)ks * NOUT);
    float4* wdst = (float4*)sW;
    for (int i = threadIdx.x; i < KSLAB * NOUT / 4; i += GEMM_THREADS)
      wdst[i] = wsrc[i];
    __syncthreads();

    // A fragments for this slab: lane holds A[m0+lm][ka], ka = k0 + 2*half + v
    v2f afrag[KSLAB / 4];
#pragma unroll
    for (int kk = 0; kk < KSLAB / 4; ++kk) {
      const int ka = ks + kk * 4 + half * 2;
      v2f a;
      a.x = xrow[ka];
      a.y = xrow[ka + 1];
      afrag[kk] = a;
    }

#pragma unroll
    for (int t = 0; t < NT; ++t) {
      const int n0 = t * 16;
#pragma unroll
      for (int kk = 0; kk < KSLAB / 4; ++kk) {
        const int ka = kk * 4 + half * 2;  // LDS-relative k
        v2f b;
        b.x = sW[ka * NOUT + n0 + lm];
        b.y = sW[(ka + 1) * NOUT + n0 + lm];
        acc[t] = __builtin_amdgcn_wmma_f32_16x16x4_f32(
            false, afrag[kk], false, b, (short)0, acc[t], false, false);
      }
    }
    __syncthreads();
  }

  if (valid) {
#pragma unroll
    for (int t = 0; t < NT; ++t) {
      const int n0 = t * 16;
#pragma unroll
      for (int r = 0; r < 8; ++r) {
        // D layout: VGPR r, lane l -> D[r + 8*(l>>4)][l&15]
        H[(size_t)(m0 + r + 8 * half) * NOUT + n0 + lm] = acc[t][r];
      }
    }
  }
}

// ---------------------------------------------------------------------------
// Degree / normalization
// ---------------------------------------------------------------------------
__global__ void k_fill1(float* __restrict__ p, int n) {
  int i = blockIdx.x * blockDim.x + threadIdx.x;
  if (i < n) p[i] = 1.0f;  // self-loop contributes 1 to every node's degree
}

__global__ void k_deg(const int* __restrict__ dst, float* __restrict__ deg, int E) {
  int i = blockIdx.x * blockDim.x + threadIdx.x;
  if (i < E) atomicAdd(&deg[dst[i]], 1.0f);
}

__global__ void k_rsqrt(float* __restrict__ p, int n) {
  int i = blockIdx.x * blockDim.x + threadIdx.x;
  if (i < n) p[i] = rsqrtf(fmaxf(p[i], 1.0f));
}

// ---------------------------------------------------------------------------
// Edge-parallel scatter: agg[dst*D + f] += dis[src]*dis[dst] * h[src*D + f]
// Thread t -> (edge = t/D, feat = t%D): coalesced gather + coalesced atomics.
// ---------------------------------------------------------------------------
template <int D>
__global__ void k_scatter(const int* __restrict__ src, const int* __restrict__ dst,
                          const float* __restrict__ dis, const float* __restrict__ h,
                          float* __restrict__ agg, long long total) {
  long long t = (long long)blockIdx.x * blockDim.x + threadIdx.x;
  if (t >= total) return;
  const int e = (int)(t / D);
  const int f = (int)(t & (D - 1));
  const int s = src[e];
  const int d = dst[e];
  const float w = dis[s] * dis[d];
  atomicAdd(&agg[(size_t)d * D + f], w * h[(size_t)s * D + f]);
}

__global__ void k_scatter1(const int* __restrict__ src, const int* __restrict__ dst,
                           const float* __restrict__ dis, const float* __restrict__ h,
                           float* __restrict__ agg, int E) {
  int e = blockIdx.x * blockDim.x + threadIdx.x;
  if (e >= E) return;
  const int s = src[e];
  const int d = dst[e];
  atomicAdd(&agg[d], dis[s] * dis[d] * h[s]);
}

// ---------------------------------------------------------------------------
// Finalize (in-place on agg): out = [relu](agg + dis[i]^2 * h + b[f])
// Self-loop term fused here instead of the scatter pass.
// ---------------------------------------------------------------------------
template <int D, bool RELU>
__global__ void k_finalize(float* __restrict__ out, const float* __restrict__ h,
                           const float* __restrict__ dis, const float* __restrict__ b,
                           long long total) {
  long long t = (long long)blockIdx.x * blockDim.x + threadIdx.x;
  if (t >= total) return;
  const int i = (int)(t / D);
  const int f = (int)(t & (D - 1));
  const float s = dis[i];
  float v = out[t] + s * s * h[t] + b[f];
  out[t] = RELU ? fmaxf(v, 0.0f) : v;
}

__global__ void k_finalize1(float* __restrict__ out, const float* __restrict__ h,
                            const float* __restrict__ dis, const float* __restrict__ b,
                            int n) {
  int i = blockIdx.x * blockDim.x + threadIdx.x;
  if (i >= n) return;
  const float s = dis[i];
  out[i] = out[i] + s * s * h[i] + b[0];
}

// ---------------------------------------------------------------------------
// Layer-3 GEMM: out-dim 1 -> one wave per node, shuffle reduction.
// ---------------------------------------------------------------------------
__global__ void k_gemm3(const float* __restrict__ h2, const float* __restrict__ W3,
                        float* __restrict__ out, int n) {
  const int wave = threadIdx.x >> 5;
  const int lane = threadIdx.x & 31;
  const int node = blockIdx.x * (blockDim.x >> 5) + wave;
  if (node >= n) return;
  const float* row = h2 + (size_t)node * 128;
  float s = 0.0f;
#pragma unroll
  for (int k = 0; k < 128; k += 32) s += row[k + lane] * W3[k + lane];
#pragma unroll
  for (int off = 16; off > 0; off >>= 1) s += __shfl_down(s, off, 32);
  if (lane == 0) out[node] = s;
}

// ---------------------------------------------------------------------------
extern "C" void kernel_launch(void* const* d_in, const int* in_sizes, int n_in,
                              void* d_out, int out_size, void* d_ws, size_t ws_size,
                              hipStream_t stream) {
  (void)n_in; (void)out_size; (void)ws_size;
  constexpr int IN = 128, HD1 = 256, HD2 = 128;

  const float* x  = (const float*)d_in[0];
  const int*   ei = (const int*)d_in[1];
  const float* W1 = (const float*)d_in[2];
  const float* b1 = (const float*)d_in[3];
  const float* W2 = (const float*)d_in[4];
  const float* b2 = (const float*)d_in[5];
  const float* W3 = (const float*)d_in[6];
  const float* b3 = (const float*)d_in[7];

  const int N = in_sizes[0] / IN;   // 50000
  const int E = in_sizes[1] / 2;    // 800000
  const int* src = ei;
  const int* dst = ei + E;

  // Output tuple (y, h1, h2) concatenated flat.
  float* y  = (float*)d_out;
  float* h1 = y + N;
  float* h2 = h1 + (size_t)N * HD1;

  // Workspace: dis[N] + pre-activation buffer (max N*256 floats).
  float* dis  = (float*)d_ws;
  float* hpre = dis + N;

  const int tb = 256;

  // --- symmetric normalization: dis = rsqrt(1 + indegree) ---
  k_fill1<<<(N + tb - 1) / tb, tb, 0, stream>>>(dis, N);
  k_deg<<<(E + tb - 1) / tb, tb, 0, stream>>>(dst, dis, E);
  k_rsqrt<<<(N + tb - 1) / tb, tb, 0, stream>>>(dis, N);

  const int mtiles  = N / 16;                                       // 3125
  const int gblocks = (mtiles + WAVES_PER_BLOCK - 1) / WAVES_PER_BLOCK;  // 625

  // --- Layer 1: h1 = relu(agg(x@W1) + b1) ---
  gemm_wmma_f32<IN, HD1><<<gblocks, GEMM_THREADS, 0, stream>>>(x, W1, hpre, mtiles);
  hipMemsetAsync(h1, 0, (size_t)N * HD1 * sizeof(float), stream);
  {
    long long total = (long long)E * HD1;
    k_scatter<HD1><<<(unsigned)((total + tb - 1) / tb), tb, 0, stream>>>(
        src, dst, dis, hpre, h1, total);
    long long ntot = (long long)N * HD1;
    k_finalize<HD1, true><<<(unsigned)((ntot + tb - 1) / tb), tb, 0, stream>>>(
        h1, hpre, dis, b1, ntot);
  }

  // --- Layer 2: h2 = relu(agg(h1@W2) + b2) ---
  gemm_wmma_f32<HD1, HD2><<<gblocks, GEMM_THREADS, 0, stream>>>(h1, W2, hpre, mtiles);
  hipMemsetAsync(h2, 0, (size_t)N * HD2 * sizeof(float), stream);
  {
    long long total = (long long)E * HD2;
    k_scatter<HD2><<<(unsigned)((total + tb - 1) / tb), tb, 0, stream>>>(
        src, dst, dis, hpre, h2, total);
    long long ntot = (long long)N * HD2;
    k_finalize<HD2, true><<<(unsigned)((ntot + tb - 1) / tb), tb, 0, stream>>>(
        h2, hpre, dis, b2, ntot);
  }

  // --- Layer 3: y = agg(h2@W3) + b3 ---
  k_gemm3<<<(N + 7) / 8, 256, 0, stream>>>(h2, W3, hpre, N);
  hipMemsetAsync(y, 0, (size_t)N * sizeof(float), stream);
  k_scatter1<<<(E + tb - 1) / tb, tb, 0, stream>>>(src, dst, dis, hpre, y, E);
  k_finalize1<<<(N + tb - 1) / tb, tb, 0, stream>>>(y, hpre, dis, b3, N);
}